// HMM_bayes_46926812676563
// MI455X (gfx1250) — compile-verified
//
#include <hip/hip_runtime.h>
#include <math.h>

// ---------------- problem dims ----------------
#define N1 16
#define N2 16
#define N3 8
#define SDIM 2048          // N1*N2*N3
#define MDIM 16
#define TSTEPS 16384
#define FWD_THREADS 256
#define PER_TH (SDIM / FWD_THREADS)   // 8

typedef __attribute__((ext_vector_type(2))) float v2f;
typedef __attribute__((ext_vector_type(8))) float v8f;

// ---------------------------------------------------------------------------
// Kernel 0: combine per-factor rates -> log(lam_c) [S x 16] and lamsum [S]
// ---------------------------------------------------------------------------
__global__ __launch_bounds__(256) void fhmm_prep(
    const float* __restrict__ lam1, const float* __restrict__ lam2,
    const float* __restrict__ lam3,
    float* __restrict__ loglam, float* __restrict__ lamsum)
{
    int s = blockIdx.x * blockDim.x + threadIdx.x;
    if (s >= SDIM) return;
    int s1 = s >> 7, s2 = (s >> 3) & 15, s3 = s & 7;
    float sum = 0.f;
    #pragma unroll
    for (int m = 0; m < MDIM; ++m) {
        float lc = lam1[s1 * MDIM + m] + lam2[s2 * MDIM + m] + lam3[s3 * MDIM + m];
        loglam[s * MDIM + m] = logf(lc);
        sum += lc;
    }
    lamsum[s] = sum;
}

// ---------------------------------------------------------------------------
// Kernel 1: em'[t,s] = dot(x[t,:], loglam[s,:]) - lamsum[s]   (T x S, f32)
// GEMM (16384x16)@(16x2048) via V_WMMA_F32_16X16X4_F32, one 16x16 tile/wave,
// K=16 -> 4 chained WMMAs. 4 waves per block.
// ---------------------------------------------------------------------------
__global__ __launch_bounds__(128) void fhmm_em_gemm(
    const int* __restrict__ x, const float* __restrict__ loglam,
    const float* __restrict__ lamsum, float* __restrict__ em)
{
    int wid  = blockIdx.x * 4 + (threadIdx.x >> 5);
    int tr   = wid >> 7;      // time tile 0..1023
    int tc   = wid & 127;     // state tile 0..127
    int lane = threadIdx.x & 31;
    int r    = lane & 15;
    int half = lane >> 4;

    const int4*   x4  = (const int4*)x;         // 4 int4 per 16-int row
    const float4* ll4 = (const float4*)loglam;  // 4 float4 per 16-f32 row

    v8f c = {};
    #pragma unroll
    for (int kk = 0; kk < 4; ++kk) {
        int4   xa = x4 [(size_t)(tr * 16 + r) * 4 + kk];
        float4 lb = ll4[(size_t)(tc * 16 + r) * 4 + kk];
        v2f a, b;
        a[0] = half ? (float)xa.z : (float)xa.x;
        a[1] = half ? (float)xa.w : (float)xa.y;
        b[0] = half ? lb.z : lb.x;
        b[1] = half ? lb.w : lb.y;
        c = __builtin_amdgcn_wmma_f32_16x16x4_f32(
                false, a, false, b, (short)0, c, false, false);
    }
    float lsv = lamsum[tc * 16 + r];
    #pragma unroll
    for (int v = 0; v < 8; ++v) {
        int row = tr * 16 + v + 8 * half;               // C/D layout: M = v + 8*half
        em[(size_t)row * SDIM + tc * 16 + r] = c[v] - lsv;
    }
}

// ---------------------------------------------------------------------------
// Kernel 2: sequential log-domain forward recurrence (1 block, 8 waves).
// alpha kept in registers; factorized exp-domain contractions in LDS;
// running max-shift accumulated in double.
// ---------------------------------------------------------------------------
__global__ __launch_bounds__(FWD_THREADS, 1) void fhmm_forward(
    const int* __restrict__ x,
    const float* __restrict__ logA1, const float* __restrict__ logA2,
    const float* __restrict__ logA3,
    const float* __restrict__ logpi1, const float* __restrict__ logpi2,
    const float* __restrict__ logpi3,
    const float* __restrict__ em, float* __restrict__ out)
{
    __shared__ float bufA[SDIM];
    __shared__ float bufB[SDIM];
    __shared__ float A1s[N1 * N1];
    __shared__ float A2s[N2 * N2];
    __shared__ float A3s[N3 * N3];
    __shared__ float  red[8];
    __shared__ double redd[8];
    __shared__ float lf[32];

    const int tid  = threadIdx.x;
    const int lane = tid & 31;
    const int wave = tid >> 5;

    // exp() the factor transition matrices; log-factorial table
    A1s[tid] = expf(logA1[tid]);
    A2s[tid] = expf(logA2[tid]);
    if (tid < N3 * N3) A3s[tid] = expf(logA3[tid]);
    if (tid < 32) lf[tid] = lgammaf((float)tid + 1.0f);
    __syncthreads();

    // cs = sum_{t,m} lgamma(x+1), accumulated in double
    double csp = 0.0;
    for (int i = tid; i < TSTEPS * MDIM; i += FWD_THREADS)
        csp += (double)lf[x[i] & 31];
    #pragma unroll
    for (int off = 16; off > 0; off >>= 1) csp += __shfl_xor(csp, off);
    if (lane == 0) redd[wave] = csp;
    __syncthreads();
    double cs = 0.0;
    for (int w = 0; w < 8; ++w) cs += redd[w];

    // init: alpha_rel = em'[0,:] + logpi ; shift = 0
    float areg[PER_TH];
    #pragma unroll
    for (int k = 0; k < PER_TH; ++k) {
        int s = tid + k * FWD_THREADS;
        int s1 = s >> 7, s2 = (s >> 3) & 15, s3 = s & 7;
        areg[k] = em[s] + logpi1[s1] + logpi2[s2] + logpi3[s3];
    }
    double shift = 0.0;

    for (int t = 1; t < TSTEPS; ++t) {
        const float* emt = em + (size_t)t * SDIM;
        __builtin_prefetch((const void*)(emt + lane * 64), 0, 1);

        // ---- max over alpha_rel (wave shuffle + one LDS pass) ----
        float pm = -INFINITY;
        #pragma unroll
        for (int k = 0; k < PER_TH; ++k) pm = fmaxf(pm, areg[k]);
        #pragma unroll
        for (int off = 16; off > 0; off >>= 1) pm = fmaxf(pm, __shfl_xor(pm, off));
        if (lane == 0) red[wave] = pm;
        __syncthreads();                                    // B1
        float mx = red[0];
        #pragma unroll
        for (int w = 1; w < 8; ++w) mx = fmaxf(mx, red[w]);

        // ---- e = exp(alpha_rel - mx) ----
        #pragma unroll
        for (int k = 0; k < PER_TH; ++k)
            bufA[tid + k * FWD_THREADS] = expf(areg[k] - mx);
        __syncthreads();                                    // B2

        // ---- contraction over p1 (stride 128) ----
        #pragma unroll
        for (int k = 0; k < PER_TH; ++k) {
            int d = tid + k * FWD_THREADS;
            int d1 = d >> 7, p23 = d & 127;
            const float* a1r = &A1s[d1 * N1];
            float acc = 0.f;
            #pragma unroll
            for (int p1 = 0; p1 < N1; ++p1)
                acc = fmaf(a1r[p1], bufA[p1 * 128 + p23], acc);
            bufB[d] = acc;
        }
        __syncthreads();                                    // B3

        // ---- contraction over p2 (stride 8) ----
        #pragma unroll
        for (int k = 0; k < PER_TH; ++k) {
            int d = tid + k * FWD_THREADS;
            int d1 = d >> 7, d2 = (d >> 3) & 15, p3 = d & 7;
            const float* a2r = &A2s[d2 * N2];
            int base = d1 * 128 + p3;
            float acc = 0.f;
            #pragma unroll
            for (int p2 = 0; p2 < N2; ++p2)
                acc = fmaf(a2r[p2], bufB[base + p2 * 8], acc);
            bufA[d] = acc;
        }
        __syncthreads();                                    // B4

        // ---- contraction over p3 (stride 1) + alpha update ----
        #pragma unroll
        for (int k = 0; k < PER_TH; ++k) {
            int d = tid + k * FWD_THREADS;
            int d12 = d >> 3, d3 = d & 7;
            const float* a3r = &A3s[d3 * N3];
            int base = d12 * 8;
            float acc = 0.f;
            #pragma unroll
            for (int p3 = 0; p3 < N3; ++p3)
                acc = fmaf(a3r[p3], bufA[base + p3], acc);
            areg[k] = emt[d] + logf(acc);
        }
        shift += (double)mx;
        // no trailing barrier needed: bufA is only rewritten after next B1
    }

    // ---- final logsumexp(alpha_rel) + shift - cs ----
    float pm = -INFINITY;
    #pragma unroll
    for (int k = 0; k < PER_TH; ++k) pm = fmaxf(pm, areg[k]);
    #pragma unroll
    for (int off = 16; off > 0; off >>= 1) pm = fmaxf(pm, __shfl_xor(pm, off));
    __syncthreads();                 // retire last red reads from main loop
    if (lane == 0) red[wave] = pm;
    __syncthreads();
    float mx = red[0];
    #pragma unroll
    for (int w = 1; w < 8; ++w) mx = fmaxf(mx, red[w]);

    double se = 0.0;
    #pragma unroll
    for (int k = 0; k < PER_TH; ++k) se += (double)expf(areg[k] - mx);
    #pragma unroll
    for (int off = 16; off > 0; off >>= 1) se += __shfl_xor(se, off);
    if (lane == 0) redd[wave] = se;
    __syncthreads();
    if (tid == 0) {
        double ssum = 0.0;
        for (int w = 0; w < 8; ++w) ssum += redd[w];
        out[0] = (float)(shift + (double)mx + log(ssum) - cs);
    }
}

// ---------------------------------------------------------------------------
extern "C" void kernel_launch(void* const* d_in, const int* in_sizes, int n_in,
                              void* d_out, int out_size, void* d_ws, size_t ws_size,
                              hipStream_t stream) {
    const int*   x      = (const int*)  d_in[0];
    const float* lam1   = (const float*)d_in[1];
    const float* lam2   = (const float*)d_in[2];
    const float* lam3   = (const float*)d_in[3];
    const float* logA1  = (const float*)d_in[4];
    const float* logA2  = (const float*)d_in[5];
    const float* logA3  = (const float*)d_in[6];
    const float* logpi1 = (const float*)d_in[7];
    const float* logpi2 = (const float*)d_in[8];
    const float* logpi3 = (const float*)d_in[9];
    float* out = (float*)d_out;

    char*  ws     = (char*)d_ws;
    float* loglam = (float*)(ws);                                   // S*16 f32
    float* lamsum = (float*)(ws + (size_t)SDIM * MDIM * 4);         // S f32
    float* em     = (float*)(ws + (size_t)SDIM * MDIM * 4
                                + (size_t)SDIM * 4);                // T*S f32 (128 MB)

    fhmm_prep<<<SDIM / 256, 256, 0, stream>>>(lam1, lam2, lam3, loglam, lamsum);

    int tiles  = (TSTEPS / 16) * (SDIM / 16);   // 131072 16x16 tiles
    fhmm_em_gemm<<<tiles / 4, 128, 0, stream>>>(x, loglam, lamsum, em);

    fhmm_forward<<<1, FWD_THREADS, 0, stream>>>(x, logA1, logA2, logA3,
                                                logpi1, logpi2, logpi3, em, out);
}